// AdaptiveEmbedding_18270790877730
// MI455X (gfx1250) — compile-verified
//
#include <hip/hip_runtime.h>

#define D_PROJ 1024
#define EMB_SCALE 32.0f

typedef __attribute__((ext_vector_type(2))) float v2f;
typedef __attribute__((ext_vector_type(8))) float v8f;

// ---------------------------------------------------------------------------
// Kernel 1: partition tokens into per-cluster lists (ws layout:
//   int counts[4]  @ offset 0
//   int lists[4][nTok] @ offset 16 bytes)
// ---------------------------------------------------------------------------
__global__ void partition_kernel(const int* __restrict__ input,
                                 int* __restrict__ counts,
                                 int* __restrict__ lists,
                                 int nTok) {
    int t = blockIdx.x * blockDim.x + threadIdx.x;
    if (t >= nTok) return;
    int v = input[t];
    int c = (v < 20000) ? 0 : (v < 60000) ? 1 : (v < 100000) ? 2 : 3;
    int pos = atomicAdd(&counts[c], 1);
    lists[c * nTok + pos] = t;
}

// ---------------------------------------------------------------------------
// Kernel 2: grouped GEMM per cluster with V_WMMA_F32_16X16X4_F32.
// Block = 128 threads (4 waves). Block tile: 16 tokens x 64 output cols.
// Each wave owns a 16x16 output tile; K-loop steps 4.
// A (16 x d_emb gathered emb rows) staged in LDS with padded stride.
// B read directly from proj (row-major [1024, d_emb]) as float2 -> b64 loads.
// ---------------------------------------------------------------------------
__global__ __launch_bounds__(128)
void adaptive_emb_gemm(const int* __restrict__ input,
                       const int* __restrict__ counts,
                       const int* __restrict__ lists,
                       const float* __restrict__ emb0, const float* __restrict__ emb1,
                       const float* __restrict__ emb2, const float* __restrict__ emb3,
                       const float* __restrict__ proj0, const float* __restrict__ proj1,
                       const float* __restrict__ proj2, const float* __restrict__ proj3,
                       float* __restrict__ out,
                       int nTok) {
    extern __shared__ float ldsA[];   // 16 * (d_emb + 4) floats

    const int c   = blockIdx.z;
    const int cnt = counts[c];
    const int t0  = blockIdx.x * 16;
    if (t0 >= cnt) return;            // uniform per block

    int left, shift;
    const float* emb;
    const float* proj;
    switch (c) {
        case 0:  left = 0;      shift = 10; emb = emb0; proj = proj0; break;
        case 1:  left = 20000;  shift = 8;  emb = emb1; proj = proj1; break;
        case 2:  left = 60000;  shift = 6;  emb = emb2; proj = proj2; break;
        default: left = 100000; shift = 4;  emb = emb3; proj = proj3; break;
    }
    const int d_emb     = 1 << shift;
    const int ldsStride = d_emb + 4;  // pad: stride % 64 banks != 0 -> conflict-free column reads

    __shared__ int toks[16];
    __shared__ int rows[16];

    const int tid = threadIdx.x;
    if (tid < 16) {
        int ti  = t0 + tid;
        int tok = lists[c * nTok + min(ti, cnt - 1)];   // clamp pads with a valid row
        toks[tid] = tok;
        rows[tid] = input[tok] - left;                  // in-range by construction
    }
    __syncthreads();

    // Cooperative stage of the 16 gathered emb rows into LDS (coalesced per row).
    const int total = d_emb << 4;     // 16 * d_emb
    for (int idx = tid; idx < total; idx += 128) {
        int r  = idx >> shift;
        int kk = idx & (d_emb - 1);
        ldsA[r * ldsStride + kk] = emb[(size_t)rows[r] * d_emb + kk];
    }
    __syncthreads();

    const int lane = tid & 31;
    const int wave = tid >> 5;
    const int n0   = blockIdx.y * 64 + wave * 16;

    // VGPR layouts for V_WMMA_F32_16X16X4_F32 (wave32):
    //  A 16x4: lane L -> M = L%16, Kbase = (L/16)*2 ; VGPR0=K+0, VGPR1=K+1
    //  B 4x16: lane L -> N = L%16, Kbase = (L/16)*2 ; VGPR0=K+0, VGPR1=K+1
    //  C 16x16: VGPR j -> M = j + 8*(L/16), N = L%16
    const int m  = lane & 15;               // A row / B,C col index
    const int kb = (lane >> 4) << 1;        // 0 or 2

    const float* aRow = ldsA + m * ldsStride;
    const float* bRow = proj + (size_t)(n0 + m) * d_emb;

    v8f acc = {};
    #pragma unroll 4
    for (int k0 = 0; k0 < d_emb; k0 += 4) {
        v2f a, b;
        a.x = aRow[k0 + kb];
        a.y = aRow[k0 + kb + 1];
        b   = *(const v2f*)(bRow + k0 + kb);   // contiguous -> global_load_b64
        acc = __builtin_amdgcn_wmma_f32_16x16x4_f32(
                  /*neg_a=*/false, a, /*neg_b=*/false, b,
                  /*c_mod=*/(short)0, acc, /*reuse_a=*/false, /*reuse_b=*/false);
    }

    // Scale and scatter D. N = m (lane%16), M = j + 8*(lane/16).
    const int nCol  = n0 + m;
    const int mBase = (lane >> 4) << 3;      // 0 or 8
    #pragma unroll
    for (int j = 0; j < 8; ++j) {
        int mm = mBase + j;
        int ti = t0 + mm;
        if (ti < cnt) {
            out[(size_t)toks[mm] * D_PROJ + nCol] = acc[j] * EMB_SCALE;
        }
    }
}

// ---------------------------------------------------------------------------
// Launch
// ---------------------------------------------------------------------------
extern "C" void kernel_launch(void* const* d_in, const int* in_sizes, int n_in,
                              void* d_out, int out_size, void* d_ws, size_t ws_size,
                              hipStream_t stream) {
    // setup_inputs order: input, emb0, proj0, emb1, proj1, emb2, proj2, emb3, proj3
    const int*   input = (const int*)  d_in[0];
    const float* emb0  = (const float*)d_in[1];
    const float* proj0 = (const float*)d_in[2];
    const float* emb1  = (const float*)d_in[3];
    const float* proj1 = (const float*)d_in[4];
    const float* emb2  = (const float*)d_in[5];
    const float* proj2 = (const float*)d_in[6];
    const float* emb3  = (const float*)d_in[7];
    const float* proj3 = (const float*)d_in[8];
    float* out = (float*)d_out;

    const int nTok = in_sizes[0];   // 4 * 2048 = 8192

    int* counts = (int*)d_ws;
    int* lists  = (int*)((char*)d_ws + 16);

    hipMemsetAsync(d_ws, 0, 16, stream);

    partition_kernel<<<(nTok + 255) / 256, 256, 0, stream>>>(input, counts, lists, nTok);

    dim3 grid((nTok + 15) / 16, D_PROJ / 64, 4);
    size_t ldsBytes = (size_t)16 * (1024 + 4) * sizeof(float);  // worst case (cluster 0)
    adaptive_emb_gemm<<<grid, 128, ldsBytes, stream>>>(
        input, counts, lists,
        emb0, emb1, emb2, emb3,
        proj0, proj1, proj2, proj3,
        out, nTok);
}